// MultiheadAttention_15083925143922
// MI455X (gfx1250) — compile-verified
//
#include <hip/hip_runtime.h>
#include <hip/hip_bf16.h>
#include <math.h>

typedef __attribute__((ext_vector_type(2))) float v2f;
typedef __attribute__((ext_vector_type(8))) float v8f;

#define WMMA_F32X4(ACC, A, B) \
    (ACC) = __builtin_amdgcn_wmma_f32_16x16x4_f32(false, (A), false, (B), (short)0, (ACC), false, false)

#define BATCH 8
#define SEQ   1024
#define DMODEL 512
#define HEADS 8
#define DK    64
#define BH    (BATCH * HEADS)          // 64
#define MROWS (BATCH * SEQ)            // 8192
#define STRIDE 1028                    // LDS strip row stride (floats): 4 banks/row -> conflict-free

// ---------------------------------------------------------------------------
// Kernel 1: fused QKV projection.  C[n, j] = sum_d Q[n,d] * W[j,d] + b[j]
// Results scattered into per-(b,h) [SEQ, DK] layout.
// ---------------------------------------------------------------------------
__global__ __launch_bounds__(256) void qkv_proj_kernel(
    const float* __restrict__ Q,
    const float* __restrict__ Wq, const float* __restrict__ bq,
    const float* __restrict__ Wk, const float* __restrict__ bk,
    const float* __restrict__ Wv, const float* __restrict__ bv,
    float* __restrict__ qd, float* __restrict__ kd, float* __restrict__ vd)
{
    const int NT_N = DMODEL / 16;          // 32
    const int NT_M = MROWS / 16;           // 512
    const int wave = threadIdx.x >> 5;
    const long tile = (long)blockIdx.x * 8 + wave;   // 3*512*32 tiles

    const int mat = (int)(tile / (NT_M * NT_N));
    const int rem = (int)(tile % (NT_M * NT_N));
    const int row0 = (rem / NT_N) * 16;
    const int col0 = (rem % NT_N) * 16;

    const float* __restrict__ W    = (mat == 0) ? Wq : (mat == 1) ? Wk : Wv;
    const float* __restrict__ bias = (mat == 0) ? bq : (mat == 1) ? bk : bv;
    float* __restrict__ dst        = (mat == 0) ? qd : (mat == 1) ? kd : vd;

    const int lane = threadIdx.x & 31;
    const int half = lane >> 4;
    const int l16  = lane & 15;

    const float* __restrict__ Arow = Q + (size_t)(row0 + l16) * DMODEL;
    const float* __restrict__ Brow = W + (size_t)(col0 + l16) * DMODEL;  // B[k][j] = W[j*512+k]

    v8f acc = {};
#pragma unroll 8
    for (int k0 = 0; k0 < DMODEL; k0 += 4) {
        const int ka = k0 + 2 * half;
        v2f a, b;
        a[0] = Arow[ka]; a[1] = Arow[ka + 1];
        b[0] = Brow[ka]; b[1] = Brow[ka + 1];
        WMMA_F32X4(acc, a, b);
    }

    const int col = col0 + l16;
    const int h = col >> 6, d = col & (DK - 1);
    const float bv_ = bias[col];
#pragma unroll
    for (int r = 0; r < 8; ++r) {
        const int row = row0 + r + 8 * half;
        const int b_ = row >> 10, s = row & (SEQ - 1);
        dst[(((size_t)b_ * HEADS + h) * SEQ + s) * DK + d] = acc[r] + bv_;
    }
}

// ---------------------------------------------------------------------------
// Kernel 2 (fused): per (b,h), a block owns a 16-row x 1024-col score strip.
//   A) scores = q @ k^T (WMMA, 8 tiles/wave kept in registers)
//   B) * scale + alibi bias, mask, block-wide row softmax (shfl + LDS)
//      -> write normalized attn to d_out AND into an LDS strip
//   C) ctx = attn_strip @ v (WMMA from LDS), K split across wave pairs
// ---------------------------------------------------------------------------
__global__ __launch_bounds__(256) void attn_fused_kernel(
    const float* __restrict__ qd, const float* __restrict__ kd,
    const float* __restrict__ vd, const unsigned char* __restrict__ mask,
    float* __restrict__ attn, float* __restrict__ ctx)
{
    __shared__ float strip[16][STRIDE];    // normalized attn strip (64.25 KB)
    __shared__ float redA[8][16];          // per-wave row max
    __shared__ float redB[8][16];          // per-wave row sum
    __shared__ float part[4][16][16];      // K-half partials for ctx

    const int wave = threadIdx.x >> 5;
    const int lane = threadIdx.x & 31;
    const int half = lane >> 4;
    const int l16  = lane & 15;

    const int bh = blockIdx.x >> 6;        // 64 row strips per (b,h)
    const int i0 = (blockIdx.x & 63) * 16;
    const int h  = bh & (HEADS - 1);

    const float* __restrict__ qh = qd + (size_t)bh * SEQ * DK;
    const float* __restrict__ kh = kd + (size_t)bh * SEQ * DK;
    const float* __restrict__ vh = vd + (size_t)bh * SEQ * DK;

    // ---- preload this lane's q-row fragment pairs (A matrix, K=64) ----
    float qa[32];
    {
        const float* qrow = qh + (size_t)(i0 + l16) * DK;
#pragma unroll
        for (int s = 0; s < 16; ++s) {
            qa[2 * s]     = qrow[4 * s + 2 * half];
            qa[2 * s + 1] = qrow[4 * s + 2 * half + 1];
        }
    }

    // ---- Stage A: 8 score tiles per wave, columns [wave*128, wave*128+127] ----
    v8f acc[8];
    {
        v8f z = {};
#pragma unroll
        for (int t = 0; t < 8; ++t) acc[t] = z;
    }
    const int jbase = wave * 128;
#pragma unroll
    for (int t = 0; t < 8; ++t) {
        const float* krow = kh + (size_t)(jbase + t * 16 + l16) * DK;
#pragma unroll
        for (int s = 0; s < 16; ++s) {
            v2f a, b;
            a[0] = qa[2 * s]; a[1] = qa[2 * s + 1];
            b[0] = krow[4 * s + 2 * half];
            b[1] = krow[4 * s + 2 * half + 1];
            WMMA_F32X4(acc[t], a, b);
        }
    }

    // ---- Stage B: scale + alibi + mask, then block-wide row softmax ----
    const float slope = exp2f(-(float)(h + 1));  // slopes[h] = 0.5^(h+1)
    const float scale = 0.125f;                  // 64^-0.5
#pragma unroll
    for (int t = 0; t < 8; ++t) {
        const int j = jbase + t * 16 + l16;
#pragma unroll
        for (int r = 0; r < 8; ++r) {
            const int i = i0 + r + 8 * half;
            const float bias = slope * -fabsf((float)((i & 15) - (j & 15)));
            float s = acc[t][r] * scale + bias;
            if (!mask[(size_t)i * SEQ + j]) s = -INFINITY;
            acc[t][r] = s;
        }
    }

    // per-lane row max over this wave's 8 tiles, reduce within 16-lane half
    float rowmax[8];
#pragma unroll
    for (int r = 0; r < 8; ++r) {
        float m = acc[0][r];
#pragma unroll
        for (int t = 1; t < 8; ++t) m = fmaxf(m, acc[t][r]);
#pragma unroll
        for (int o = 8; o > 0; o >>= 1) m = fmaxf(m, __shfl_xor(m, o, 32));
        rowmax[r] = m;
        redA[wave][8 * half + r] = m;      // all lanes of the half write same value
    }
    __syncthreads();
    float blockmax[8];
#pragma unroll
    for (int r = 0; r < 8; ++r) {
        float m = redA[0][8 * half + r];
#pragma unroll
        for (int w = 1; w < 8; ++w) m = fmaxf(m, redA[w][8 * half + r]);
        blockmax[r] = m;
    }

    // exp and row sums
    float rowsum[8];
#pragma unroll
    for (int r = 0; r < 8; ++r) {
        float s = 0.0f;
#pragma unroll
        for (int t = 0; t < 8; ++t) {
            const float e = expf(acc[t][r] - blockmax[r]);
            acc[t][r] = e;
            s += e;
        }
#pragma unroll
        for (int o = 8; o > 0; o >>= 1) s += __shfl_xor(s, o, 32);
        rowsum[r] = s;
        redB[wave][8 * half + r] = s;
    }
    __syncthreads();
    float inv[8];
#pragma unroll
    for (int r = 0; r < 8; ++r) {
        float s = redB[0][8 * half + r];
#pragma unroll
        for (int w = 1; w < 8; ++w) s += redB[w][8 * half + r];
        inv[r] = 1.0f / s;
    }

    // normalize; write attn output once, and stage strip in LDS for Stage C
    float* __restrict__ outp = attn + (size_t)bh * SEQ * SEQ;
#pragma unroll
    for (int t = 0; t < 8; ++t) {
        const int j = jbase + t * 16 + l16;
#pragma unroll
        for (int r = 0; r < 8; ++r) {
            const int i = i0 + r + 8 * half;
            const float v = acc[t][r] * inv[r];
            outp[(size_t)i * SEQ + j] = v;
            strip[r + 8 * half][j] = v;
        }
    }
    __syncthreads();

    // ---- Stage C: ctx tile = strip @ v ; wave -> (col tile, K half) ----
    const int ctile = wave & 3;            // ctx columns [ctile*16, +15]
    const int khalf = wave >> 2;           // K range [khalf*512, +511]
    const int c0 = ctile * 16;
    v8f cacc = {};
#pragma unroll 8
    for (int k0 = 0; k0 < 512; k0 += 4) {
        const int ka = khalf * 512 + k0 + 2 * half;
        v2f a, b;
        a[0] = strip[l16][ka];
        a[1] = strip[l16][ka + 1];
        b[0] = vh[(size_t)ka * DK + c0 + l16];
        b[1] = vh[(size_t)(ka + 1) * DK + c0 + l16];
        WMMA_F32X4(cacc, a, b);
    }
    if (khalf == 1) {
#pragma unroll
        for (int r = 0; r < 8; ++r) part[ctile][r + 8 * half][l16] = cacc[r];
    }
    __syncthreads();
    if (khalf == 0) {
        const int b_ = bh >> 3;
#pragma unroll
        for (int r = 0; r < 8; ++r) {
            const int i = i0 + r + 8 * half;
            ctx[((size_t)b_ * SEQ + i) * DMODEL + h * DK + c0 + l16] =
                cacc[r] + part[ctile][r + 8 * half][l16];
        }
    }
}

// ---------------------------------------------------------------------------
// Kernel 3: out = ctx @ Wo^T + bo.  M=8192, N=512, K=512.
// ---------------------------------------------------------------------------
__global__ __launch_bounds__(256) void oproj_kernel(
    const float* __restrict__ ctx, const float* __restrict__ Wo,
    const float* __restrict__ bo, float* __restrict__ out)
{
    const int NT_N = DMODEL / 16;          // 32
    const int wave = threadIdx.x >> 5;
    const long tile = (long)blockIdx.x * 8 + wave;   // 512*32 tiles
    const int row0 = (int)(tile / NT_N) * 16;
    const int col0 = (int)(tile % NT_N) * 16;

    const int lane = threadIdx.x & 31;
    const int half = lane >> 4;
    const int l16  = lane & 15;

    const float* __restrict__ Arow = ctx + (size_t)(row0 + l16) * DMODEL;
    const float* __restrict__ Brow = Wo + (size_t)(col0 + l16) * DMODEL;

    v8f acc = {};
#pragma unroll 8
    for (int k0 = 0; k0 < DMODEL; k0 += 4) {
        const int ka = k0 + 2 * half;
        v2f a, b;
        a[0] = Arow[ka]; a[1] = Arow[ka + 1];
        b[0] = Brow[ka]; b[1] = Brow[ka + 1];
        WMMA_F32X4(acc, a, b);
    }

    const int col = col0 + l16;
    const float bo_ = bo[col];
#pragma unroll
    for (int r = 0; r < 8; ++r) {
        const int row = row0 + r + 8 * half;
        out[(size_t)row * DMODEL + col] = acc[r] + bo_;
    }
}

// ---------------------------------------------------------------------------
// Launch
// ---------------------------------------------------------------------------
extern "C" void kernel_launch(void* const* d_in, const int* in_sizes, int n_in,
                              void* d_out, int out_size, void* d_ws, size_t ws_size,
                              hipStream_t stream)
{
    const float* Q  = (const float*)d_in[0];
    const float* Wq = (const float*)d_in[1];
    const float* bq = (const float*)d_in[2];
    const float* Wk = (const float*)d_in[3];
    const float* bk = (const float*)d_in[4];
    const float* Wv = (const float*)d_in[5];
    const float* bv = (const float*)d_in[6];
    const float* Wo = (const float*)d_in[7];
    const float* bo = (const float*)d_in[8];
    const unsigned char* mask = (const unsigned char*)d_in[9];

    float* out  = (float*)d_out;                                   // [8,1024,512]
    float* attn = (float*)d_out + (size_t)MROWS * DMODEL;          // [8,8,1024,1024]

    const size_t headElems = (size_t)BH * SEQ * DK;                // 4M floats
    float* qd  = (float*)d_ws;
    float* kd  = qd + headElems;
    float* vd  = kd + headElems;
    float* ctx = vd + headElems;                                   // [8192, 512]

    dim3 blk(256);

    // 1) QKV projections: 3 * 512 * 32 tiles / 8 waves = 6144 blocks
    qkv_proj_kernel<<<6144, blk, 0, stream>>>(Q, Wq, bq, Wk, bk, Wv, bv, qd, kd, vd);

    // 2) Fused scores+softmax+attnV: 64 (b,h) * 64 strips = 4096 blocks
    attn_fused_kernel<<<4096, blk, 0, stream>>>(qd, kd, vd, mask, attn, ctx);

    // 3) Output projection: 512 * 32 tiles / 8 = 2048 blocks
    oproj_kernel<<<2048, blk, 0, stream>>>(ctx, Wo, bo, out);
}